// QnnFormerVQC_61512521613697
// MI455X (gfx1250) — compile-verified
//
#include <hip/hip_runtime.h>
#include <hip/hip_bf16.h>

#define NQ 4
#define NL 3

// ---------------- complex helpers for the tiny matrix-build kernel ----------
struct cplx { float re, im; };
__device__ __forceinline__ cplx cmul(cplx a, cplx b){ return {a.re*b.re - a.im*b.im, a.re*b.im + a.im*b.re}; }
__device__ __forceinline__ cplx cadd(cplx a, cplx b){ return {a.re+b.re, a.im+b.im}; }

__device__ __forceinline__ void apply1(cplx* st, int mask, cplx m00, cplx m01, cplx m10, cplx m11){
  for (int n = 0; n < 16; ++n){
    if (n & mask) continue;
    cplx a = st[n], b = st[n|mask];
    st[n]      = cadd(cmul(m00,a), cmul(m01,b));
    st[n|mask] = cadd(cmul(m10,a), cmul(m11,b));
  }
}
__device__ __forceinline__ void applyC(cplx* st, int cmask, int tmask, cplx m00, cplx m01, cplx m10, cplx m11){
  for (int n = 0; n < 16; ++n){
    if (!(n & cmask) || (n & tmask)) continue;
    cplx a = st[n], b = st[n|tmask];
    st[n]       = cadd(cmul(m00,a), cmul(m01,b));
    st[n|tmask] = cadd(cmul(m10,a), cmul(m11,b));
  }
}

// Build the batch-independent 16x16 circuit unitary M (layers only; the data
// encoding is handled per-batch in the WMMA kernel). Column `col` = circuit
// applied to basis state e_col. Qubit q <-> bit (8 >> q) of the state index.
__global__ void vqc_build_matrix(const float* __restrict__ W, const float* __restrict__ E,
                                 const float* __restrict__ G, const float* __restrict__ Bt,
                                 float* __restrict__ Mr, float* __restrict__ Mi)
{
  const int col = threadIdx.x;
  if (col >= 16) return;
  cplx st[16];
  for (int j = 0; j < 16; ++j) st[j] = { (j == col) ? 1.f : 0.f, 0.f };

  const float r = 0.70710678118654752f;
  const cplx H00{ r,0}, H01{r,0}, H10{r,0}, H11{-r,0};
  const cplx X00{ 0,0}, X01{1,0}, X10{1,0}, X11{ 0,0};

  for (int l = 0; l < NL; ++l){
    for (int q = 0; q < NQ; ++q){
      float th = W[l*NQ + q];
      float c = cosf(0.5f*th), s = sinf(0.5f*th);
      int m = 8 >> q;
      apply1(st, m, cplx{c,0}, cplx{0,-s}, cplx{0,-s}, cplx{c,0});   // RX
      apply1(st, m, cplx{c,0}, cplx{-s,0}, cplx{s,0},  cplx{c,0});   // RY
      apply1(st, m, cplx{c,-s}, cplx{0,0}, cplx{0,0},  cplx{c,s});   // RZ
    }
    for (int i = 0; i < NQ; ++i){                                    // CRX ring
      float th = E[l*NQ + i];
      float c = cosf(0.5f*th), s = sinf(0.5f*th);
      applyC(st, 8 >> i, 8 >> ((i+1)&3), cplx{c,0}, cplx{0,-s}, cplx{0,-s}, cplx{c,0});
    }
    // Grover diffusion: H all, X all, H(q3), MCX(0,1,2->3), H(q3), X all, H all
    for (int q = 0; q < NQ; ++q) apply1(st, 8 >> q, H00,H01,H10,H11);
    for (int q = 0; q < NQ; ++q) apply1(st, 8 >> q, X00,X01,X10,X11);
    apply1(st, 1, H00,H01,H10,H11);
    { cplx t = st[14]; st[14] = st[15]; st[15] = t; }                // MCX: swap |1110>,|1111>
    apply1(st, 1, H00,H01,H10,H11);
    for (int q = 0; q < NQ; ++q) apply1(st, 8 >> q, X00,X01,X10,X11);
    for (int q = 0; q < NQ; ++q) apply1(st, 8 >> q, H00,H01,H10,H11);
    // CX - RZ(gamma) - CX chain
    float g = G[l]; float cg = cosf(0.5f*g), sg = sinf(0.5f*g);
    for (int i = 0; i < NQ-1; ++i){
      applyC(st, 8 >> i, 8 >> (i+1), X00,X01,X10,X11);
      apply1(st, 8 >> (i+1), cplx{cg,-sg}, cplx{0,0}, cplx{0,0}, cplx{cg,sg});
      applyC(st, 8 >> i, 8 >> (i+1), X00,X01,X10,X11);
    }
    // RX(beta) mixer
    float b = Bt[l]; float cb = cosf(0.5f*b), sb = sinf(0.5f*b);
    for (int q = 0; q < NQ; ++q)
      apply1(st, 8 >> q, cplx{cb,0}, cplx{0,-sb}, cplx{0,-sb}, cplx{cb,0});
  }
  for (int j = 0; j < 16; ++j){ Mr[j*16 + col] = st[j].re; Mi[j*16 + col] = st[j].im; }
}

// ---------------- main batched kernel: encode + (Mr,Mi) matvec via WMMA -----
typedef __attribute__((ext_vector_type(2))) float v2f;
typedef __attribute__((ext_vector_type(8))) float v8f;

__global__ __launch_bounds__(256) void vqc_wmma_kernel(const float* __restrict__ x,
    const float* __restrict__ Mr, const float* __restrict__ Mi,
    float* __restrict__ out, int Bn, int tiles)
{
  __shared__ float sRe[8][16][17];   // padded: stride 17 kills 4-way bank conflicts
  __shared__ float sIm[8][16][17];

  const int lane = threadIdx.x & 31;
  const int wv   = threadIdx.x >> 5;
  int tile = blockIdx.x * 8 + wv;
  const bool valid = tile < tiles;
  if (!valid) tile = 0;              // branchless clamp; EXEC stays all-ones for WMMA
  const int rowBase = tile * 16;
  const int m  = lane & 15;          // batch row within tile / B-matrix column
  const int hi = lane >> 4;          // K-half selector per the f32 A/B layouts
  const int row = min(rowBase + m, Bn - 1);

  // Data-encoding: one b128 load, per-qubit (cos,sin) of x/2.
  const float4 xv = ((const float4*)x)[row];
  float c[4], s[4];
  c[0] = cosf(0.5f*xv.x); s[0] = sinf(0.5f*xv.x);
  c[1] = cosf(0.5f*xv.y); s[1] = sinf(0.5f*xv.y);
  c[2] = cosf(0.5f*xv.z); s[2] = sinf(0.5f*xv.z);
  c[3] = cosf(0.5f*xv.w); s[3] = sinf(0.5f*xv.w);

  // A-operand for lane: k = 4*kk + 2*hi + b0. Only bit1 of k (qubit 2's
  // factor) is lane-dependent -> ONE cndmask, everything else is
  // constant-index products. (Avoids the compiler's dynamic-extract
  // cmp/cndmask trees seen in the previous build.)
  const float f2 = hi ? s[2] : c[2];
  float a0[4], a1[4];
  #pragma unroll
  for (int kk = 0; kk < 4; ++kk){
    const float g = ((kk&2) ? s[0] : c[0]) * ((kk&1) ? s[1] : c[1]) * f2;
    a0[kk] = g * c[3];   // b0 = 0
    a1[kk] = g * s[3];   // b0 = 1
  }

  // psi = M * p  as two real 16x16 tiles: D[row][n] = sum_k p[row][k] * M[n][k]
  v8f accR = {0,0,0,0,0,0,0,0};
  v8f accI = {0,0,0,0,0,0,0,0};
  #pragma unroll
  for (int kk = 0; kk < 4; ++kk){
    const int k0 = kk*4 + hi*2;
    v2f a;  a[0]  = a0[kk];          a[1]  = a1[kk];
    v2f br; br[0] = Mr[m*16 + k0];   br[1] = Mr[m*16 + k0 + 1];
    v2f bi; bi[0] = Mi[m*16 + k0];   bi[1] = Mi[m*16 + k0 + 1];
    accR = __builtin_amdgcn_wmma_f32_16x16x4_f32(false, a, false, br, (short)0, accR, false, false);
    accI = __builtin_amdgcn_wmma_f32_16x16x4_f32(false, a, false, bi, (short)0, accI, false, false);
  }

  // D layout: VGPR rr -> batch row rr + 8*hi, state index n = lane&15.
  #pragma unroll
  for (int rr = 0; rr < 8; ++rr){
    const int rw = rr + hi*8;
    sRe[wv][rw][m] = accR[rr];
    sIm[wv][rw][m] = accI[rr];
  }
  __syncthreads();

  // Observables. Both lane-halves read row (lane&15) via LDS broadcast and
  // compute all 12 values; half 0 stores outputs 0..5 (z0..z3,x0,x1), half 1
  // stores 6..11 (x2,x3,y0..y3) -> the wave writes a contiguous 768B region
  // with 3 coalesced global_store_b64 per lane.
  if (valid){
    float re[16], im[16];
    #pragma unroll
    for (int n = 0; n < 16; ++n){ re[n] = sRe[wv][m][n]; im[n] = sIm[wv][m][n]; }
    float o[12];
    #pragma unroll
    for (int q = 0; q < 4; ++q){
      const int msk = 8 >> q;
      float zq = 0.f, xq = 0.f, yq = 0.f;
      #pragma unroll
      for (int n = 0; n < 16; ++n){
        if (n & msk) continue;
        const int d = n | msk;
        zq += re[n]*re[n] + im[n]*im[n] - re[d]*re[d] - im[d]*im[d];
        xq += re[n]*re[d] + im[n]*im[d];
        yq += re[n]*im[d] - im[n]*re[d];
      }
      o[q]     = zq;
      o[4 + q] = 2.f * xq;
      o[8 + q] = 2.f * yq;
    }
    const int b = rowBase + m;
    if (b < Bn){
      float2 w0, w1, w2;                       // constant-index cndmask selects
      w0.x = hi ? o[6]  : o[0];  w0.y = hi ? o[7]  : o[1];
      w1.x = hi ? o[8]  : o[2];  w1.y = hi ? o[9]  : o[3];
      w2.x = hi ? o[10] : o[4];  w2.y = hi ? o[11] : o[5];
      float2* dst = (float2*)(out + b*12 + hi*6);   // 8B-aligned
      dst[0] = w0; dst[1] = w1; dst[2] = w2;
    }
  }
}

extern "C" void kernel_launch(void* const* d_in, const int* in_sizes, int n_in,
                              void* d_out, int out_size, void* d_ws, size_t ws_size,
                              hipStream_t stream)
{
  const float* x  = (const float*)d_in[0];   // (B, 4)
  const float* W  = (const float*)d_in[1];   // (3, 4)
  const float* E  = (const float*)d_in[2];   // (3, 4)
  const float* G  = (const float*)d_in[3];   // (3,)
  const float* Bt = (const float*)d_in[4];   // (3,)
  float* out = (float*)d_out;

  float* Mr = (float*)d_ws;                  // 256 floats
  float* Mi = Mr + 256;                      // 256 floats

  vqc_build_matrix<<<1, 16, 0, stream>>>(W, E, G, Bt, Mr, Mi);

  const int Bn     = in_sizes[0] / 4;        // 262144
  const int tiles  = (Bn + 15) / 16;         // 16384 waves
  const int blocks = (tiles + 7) / 8;        // 8 waves / 256-thread block
  vqc_wmma_kernel<<<blocks, 256, 0, stream>>>(x, Mr, Mi, out, Bn, tiles);
}